// PointNet2SemSeg_7954279432405
// MI455X (gfx1250) — compile-verified
//
#include <hip/hip_runtime.h>
#include <hip/hip_bf16.h>

// ---------------------------------------------------------------------------
// PointNet++ sem-seg forward for MI455X (gfx1250, wave32, WMMA).
// All GEMMs (the per-point MLPs) run through V_WMMA_F32_16X16X4_F32 so the
// matrix math stays in fp32 (reference precision) while using the matrix
// pipes. FPS keeps the whole per-batch point cloud in LDS (<=64KB of the
// 320KB WGP LDS).
// ---------------------------------------------------------------------------

typedef __attribute__((ext_vector_type(2))) float v2f;
typedef __attribute__((ext_vector_type(8))) float v8f;

#define BB 8            // batch
#define NPTS 4096       // input points per batch

// ---------------------------- kernels --------------------------------------

// (B,3,N) -> (B,N,3)
__global__ void k_transpose_b3n(const float* __restrict__ src,
                                float* __restrict__ dst, int N) {
  int idx = blockIdx.x * blockDim.x + threadIdx.x;
  int total = BB * N;
  if (idx >= total) return;
  int b = idx / N, n = idx % N;
  const float* s = src + (size_t)b * 3 * N;
  float* d = dst + (size_t)idx * 3;
  d[0] = s[n];
  d[1] = s[N + n];
  d[2] = s[2 * N + n];
}

// Farthest point sampling. One workgroup per batch; points + running min-dist
// live entirely in LDS. Matches the reference: idx[0] is always 0, idx[t+1] is
// argmax of the updated min-distance field (first-index tie break).
__global__ void __launch_bounds__(256) k_fps(const float* __restrict__ xyz,
                                             int* __restrict__ fidx,
                                             int N, int npoint) {
  __shared__ float sx[NPTS], sy[NPTS], sz[NPTS], sd[NPTS];
  __shared__ float rv[256];
  __shared__ int ri[256];
  __shared__ int s_far;
  int b = blockIdx.x;
  int t = threadIdx.x;
  const float* base = xyz + (size_t)b * N * 3;
  for (int i = t; i < N; i += 256) {
    sx[i] = base[3 * i + 0];
    sy[i] = base[3 * i + 1];
    sz[i] = base[3 * i + 2];
    sd[i] = 1e10f;
  }
  if (t == 0) s_far = 0;
  __syncthreads();
  for (int it = 0; it < npoint; ++it) {
    int far = s_far;
    if (t == 0) fidx[b * npoint + it] = far;
    float cx = sx[far], cy = sy[far], cz = sz[far];
    float bestv = -1.0f;
    int besti = 0;
    for (int i = t; i < N; i += 256) {
      float dx = sx[i] - cx, dy = sy[i] - cy, dz = sz[i] - cz;
      float d = dx * dx + dy * dy + dz * dz;
      float nd = sd[i] < d ? sd[i] : d;
      sd[i] = nd;
      if (nd > bestv) { bestv = nd; besti = i; }
    }
    rv[t] = bestv;
    ri[t] = besti;
    __syncthreads();
    for (int s = 128; s > 0; s >>= 1) {
      if (t < s) {
        if (rv[t + s] > rv[t] ||
            (rv[t + s] == rv[t] && ri[t + s] < ri[t])) {
          rv[t] = rv[t + s];
          ri[t] = ri[t + s];
        }
      }
      __syncthreads();
    }
    if (t == 0) s_far = ri[0];
    __syncthreads();
  }
}

__global__ void k_gather_xyz(const float* __restrict__ xyz,
                             const int* __restrict__ fidx,
                             float* __restrict__ out, int N, int npoint) {
  int idx = blockIdx.x * blockDim.x + threadIdx.x;
  int total = BB * npoint;
  if (idx >= total) return;
  int b = idx / npoint;
  int j = fidx[idx];
  const float* s = xyz + ((size_t)b * N + j) * 3;
  float* d = out + (size_t)idx * 3;
  d[0] = s[0]; d[1] = s[1]; d[2] = s[2];
}

// Ball query: first `ns` indices (ascending) with d^2<=r^2, padded with the
// first hit (reference semantics of sort + where(idx==N, first, idx)).
__global__ void k_query_ball(const float* __restrict__ xyz,
                             const float* __restrict__ new_xyz,
                             int* __restrict__ gidx,
                             int N, int np, int ns, float r2) {
  int idx = blockIdx.x * blockDim.x + threadIdx.x;
  int total = BB * np;
  if (idx >= total) return;
  int b = idx / np;
  const float* c = new_xyz + (size_t)idx * 3;
  float cx = c[0], cy = c[1], cz = c[2];
  const float* base = xyz + (size_t)b * N * 3;
  int* out = gidx + (size_t)idx * ns;
  int cnt = 0, first = 0;
  bool have = false;
  for (int j = 0; j < N; ++j) {
    float dx = base[3 * j + 0] - cx;
    float dy = base[3 * j + 1] - cy;
    float dz = base[3 * j + 2] - cz;
    float d = dx * dx + dy * dy + dz * dz;
    if (d <= r2) {
      if (!have) { first = j; have = true; }
      if (cnt < ns) out[cnt++] = j;
    }
  }
  while (cnt < ns) out[cnt++] = first;
}

// Build grouped feature rows [B*np*ns, 3+C]: centered xyz then gathered feats.
__global__ void k_group(const float* __restrict__ xyz,
                        const float* __restrict__ feat,
                        const float* __restrict__ new_xyz,
                        const int* __restrict__ gidx,
                        float* __restrict__ out,
                        int N, int np, int ns, int C) {
  int Cg = 3 + C;
  size_t total = (size_t)BB * np * ns * Cg;
  size_t idx = (size_t)blockIdx.x * blockDim.x + threadIdx.x;
  if (idx >= total) return;
  size_t row = idx / Cg;
  int c = (int)(idx % Cg);
  int b = (int)(row / ((size_t)np * ns));
  int m = (int)((row / ns) % np);
  int j = gidx[row];
  float v;
  if (c < 3) {
    v = xyz[((size_t)b * N + j) * 3 + c] -
        new_xyz[((size_t)b * np + m) * 3 + c];
  } else {
    v = feat[((size_t)b * N + j) * C + (c - 3)];
  }
  out[idx] = v;
}

__global__ void k_maxpool(const float* __restrict__ act,
                          float* __restrict__ out,
                          int rows, int ns, int C) {
  size_t total = (size_t)rows * C;
  size_t idx = (size_t)blockIdx.x * blockDim.x + threadIdx.x;
  if (idx >= total) return;
  size_t r = idx / C;
  int c = (int)(idx % C);
  float m = -3.4e38f;
  for (int s = 0; s < ns; ++s) {
    float v = act[((size_t)r * ns + s) * C + c];
    m = v > m ? v : m;
  }
  out[idx] = m;
}

// 3 nearest neighbours of each xyz1 point among xyz2, with inverse-distance
// weights (reference: w = 1/(d+1e-8), normalized).
__global__ void k_three_nn(const float* __restrict__ xyz1,
                           const float* __restrict__ xyz2,
                           int* __restrict__ idx3, float* __restrict__ w3,
                           int N1, int N2) {
  int idx = blockIdx.x * blockDim.x + threadIdx.x;
  int total = BB * N1;
  if (idx >= total) return;
  int b = idx / N1;
  const float* p = xyz1 + (size_t)idx * 3;
  float px = p[0], py = p[1], pz = p[2];
  const float* base = xyz2 + (size_t)b * N2 * 3;
  float d0 = 1e30f, d1 = 1e30f, d2 = 1e30f;
  int i0 = 0, i1 = 0, i2 = 0;
  for (int j = 0; j < N2; ++j) {
    float dx = base[3 * j + 0] - px;
    float dy = base[3 * j + 1] - py;
    float dz = base[3 * j + 2] - pz;
    float d = dx * dx + dy * dy + dz * dz;
    if (d < d0) {
      d2 = d1; i2 = i1; d1 = d0; i1 = i0; d0 = d; i0 = j;
    } else if (d < d1) {
      d2 = d1; i2 = i1; d1 = d; i1 = j;
    } else if (d < d2) {
      d2 = d; i2 = j;
    }
  }
  float w0 = 1.0f / (d0 + 1e-8f);
  float w1 = 1.0f / (d1 + 1e-8f);
  float w2 = 1.0f / (d2 + 1e-8f);
  float ws = w0 + w1 + w2;
  idx3[idx * 3 + 0] = i0;
  idx3[idx * 3 + 1] = i1;
  idx3[idx * 3 + 2] = i2;
  w3[idx * 3 + 0] = w0 / ws;
  w3[idx * 3 + 1] = w1 / ws;
  w3[idx * 3 + 2] = w2 / ws;
}

// concat(points1, sum_k points2[idx_k]*w_k) -> [B*N1, C1+C2]
__global__ void k_concat_interp(const float* __restrict__ p1,
                                const float* __restrict__ p2,
                                const int* __restrict__ idx3,
                                const float* __restrict__ w3,
                                float* __restrict__ out,
                                int N1, int N2, int C1, int C2) {
  int Ct = C1 + C2;
  size_t total = (size_t)BB * N1 * Ct;
  size_t idx = (size_t)blockIdx.x * blockDim.x + threadIdx.x;
  if (idx >= total) return;
  size_t row = idx / Ct;
  int c = (int)(idx % Ct);
  float v;
  if (c < C1) {
    v = p1[row * C1 + c];
  } else {
    int b = (int)(row / N1);
    int cc = c - C1;
    const float* base = p2 + (size_t)b * N2 * C2;
    v = base[(size_t)idx3[row * 3 + 0] * C2 + cc] * w3[row * 3 + 0] +
        base[(size_t)idx3[row * 3 + 1] * C2 + cc] * w3[row * 3 + 1] +
        base[(size_t)idx3[row * 3 + 2] * C2 + cc] * w3[row * 3 + 2];
  }
  out[idx] = v;
}

// ---------------------------------------------------------------------------
// fp32 WMMA GEMM + fused bias / batchnorm-affine / ReLU epilogue.
// Y[M,N] = relu( (X[M,K] @ W[K,N] + b) * g/sqrt(1+eps) + be )
//
// Contracts (guaranteed by the launcher for this network):
//   * M is a multiple of 64  -> no row guards anywhere.
//   * N guard handled by clamped column indices + masked stores (branchless).
//   * K tail (K%4 != 0) handled by one masked step: A fragment zeroed via
//     selects after unconditional clamped loads; EXEC stays all-1s for WMMA.
//
// 4 waves per block, each wave owns one 16-row tile and produces a 16x64
// strip: the A fragment is reused across 4 V_WMMA_F32_16X16X4_F32 per K-step.
//
// V_WMMA_F32_16X16X4_F32 fragment layouts (ISA 7.12.2):
//   A 16x4 : lane holds M=lane&15, K = 2*(lane>>4)+{0,1}  (v2f)
//   B 4x16 : lane holds N=lane&15, K = 2*(lane>>4)+{vgpr} (v2f)
//   C 16x16: VGPR i -> M = i + 8*(lane>>4), N = lane&15   (v8f)
// ---------------------------------------------------------------------------
__global__ void __launch_bounds__(128) k_gemm_bn_relu(
    const float* __restrict__ X, const float* __restrict__ Wm,
    const float* __restrict__ bias, const float* __restrict__ g,
    const float* __restrict__ be, float* __restrict__ Y,
    int M, int K, int N, int do_bn_relu) {
  const int lane = threadIdx.x & 31;
  const int wave = threadIdx.x >> 5;
  const int half = lane >> 4;
  const int l16 = lane & 15;
  const int tm = (blockIdx.y * 4 + wave) * 16;   // M % 64 == 0 guaranteed
  const int tn0 = blockIdx.x * 64;
  const int m = tm + l16;
  const float* __restrict__ Xrow = X + (size_t)m * K;

  // Clamped column index per tile: loads always in-bounds, stores masked.
  int nc[4];
#pragma unroll
  for (int t = 0; t < 4; ++t) {
    const int n = tn0 + t * 16 + l16;
    nc[t] = n < N ? n : N - 1;
  }

  v8f acc[4] = {v8f{}, v8f{}, v8f{}, v8f{}};

  const int Kmain = K & ~3;
  int k0 = 0;
  for (; k0 < Kmain; k0 += 4) {
    const int ka = k0 + 2 * half;
    v2f a;
    a.x = Xrow[ka];
    a.y = Xrow[ka + 1];
    __builtin_prefetch(Xrow + k0 + 16, 0, 0);   // speculative; dropped if OOB
    const float* __restrict__ w0 = Wm + (size_t)ka * N;
    const float* __restrict__ w1 = w0 + N;
#pragma unroll
    for (int t = 0; t < 4; ++t) {
      v2f bv;
      bv.x = w0[nc[t]];
      bv.y = w1[nc[t]];
      acc[t] = __builtin_amdgcn_wmma_f32_16x16x4_f32(
          false, a, false, bv, (short)0, acc[t], false, false);
    }
  }
  if (k0 < K) {  // K tail (1..3 leftover): uniform branch, branchless lanes
    const int ka = k0 + 2 * half;
    const int i0 = ka < K ? ka : K - 1;       // clamped, always valid
    const int i1 = ka + 1 < K ? ka + 1 : K - 1;
    const float v0 = Xrow[i0];
    const float v1 = Xrow[i1];
    v2f a;
    a.x = ka < K ? v0 : 0.0f;       // zero A kills garbage B rows
    a.y = ka + 1 < K ? v1 : 0.0f;
    const float* __restrict__ w0 = Wm + (size_t)i0 * N;
    const float* __restrict__ w1 = Wm + (size_t)i1 * N;
#pragma unroll
    for (int t = 0; t < 4; ++t) {
      v2f bv;
      bv.x = w0[nc[t]];
      bv.y = w1[nc[t]];
      acc[t] = __builtin_amdgcn_wmma_f32_16x16x4_f32(
          false, a, false, bv, (short)0, acc[t], false, false);
    }
  }

#pragma unroll
  for (int t = 0; t < 4; ++t) {
    const int n = tn0 + t * 16 + l16;
    if (n >= N) continue;   // store mask only (after all WMMAs)
    const float bi = bias ? bias[n] : 0.0f;
    float sc = 1.0f, bt = 0.0f;
    if (do_bn_relu) {
      sc = g[n] * rsqrtf(1.0f + 1e-5f);
      bt = be[n];
    }
#pragma unroll
    for (int i = 0; i < 8; ++i) {
      const int row = tm + i + 8 * half;
      float v = acc[t][i] + bi;
      if (do_bn_relu) {
        v = v * sc + bt;
        v = v > 0.0f ? v : 0.0f;
      }
      Y[(size_t)row * N + n] = v;
    }
  }
}

// ---------------------------- host side ------------------------------------

namespace {

struct Layer {
  const float *W, *b, *g, *be;
  int cin, cout;
};

inline void launch_gemm(hipStream_t s, const float* X, const Layer& L,
                        float* Y, int M, int bnrelu) {
  // M must be a multiple of 64 (true for every stage of this network).
  dim3 grid((L.cout + 63) / 64, M / 64);
  k_gemm_bn_relu<<<grid, 128, 0, s>>>(X, L.W, L.b, L.g, L.be, Y, M, L.cin,
                                      L.cout, bnrelu);
}

// Run an MLP chain ping-ponging between bufA/bufB; last layer optionally
// writes straight to final_out. Returns pointer holding the result.
const float* run_mlp(hipStream_t s, const float* in, int M, const Layer* Ls,
                     int nL, float* bufA, float* bufB, float* final_out) {
  const float* cur = in;
  for (int i = 0; i < nL; ++i) {
    float* out = (i == nL - 1 && final_out)
                     ? final_out
                     : ((cur == bufA) ? bufB : bufA);
    launch_gemm(s, cur, Ls[i], out, M, /*bnrelu=*/1);
    cur = out;
  }
  return cur;
}

void run_sa(hipStream_t s, const float* xyz_in, int Nin, const float* feat_in,
            int Cin, int np, float radius, int ns, const Layer* Ls, int nL,
            float* xyz_out, float* feat_out, int* fidx, int* gidx,
            float* bufA, float* bufB) {
  k_fps<<<BB, 256, 0, s>>>(xyz_in, fidx, Nin, np);
  int tot = BB * np;
  k_gather_xyz<<<(tot + 255) / 256, 256, 0, s>>>(xyz_in, fidx, xyz_out, Nin, np);
  k_query_ball<<<(tot + 255) / 256, 256, 0, s>>>(xyz_in, xyz_out, gidx, Nin,
                                                 np, ns, radius * radius);
  size_t ge = (size_t)BB * np * ns * (3 + Cin);
  k_group<<<(unsigned)((ge + 255) / 256), 256, 0, s>>>(
      xyz_in, feat_in, xyz_out, gidx, bufA, Nin, np, ns, Cin);
  int Mrows = BB * np * ns;
  const float* act = run_mlp(s, bufA, Mrows, Ls, nL, bufA, bufB, nullptr);
  int Cout = Ls[nL - 1].cout;
  size_t pt = (size_t)BB * np * Cout;
  k_maxpool<<<(unsigned)((pt + 255) / 256), 256, 0, s>>>(act, feat_out,
                                                         BB * np, ns, Cout);
}

void run_fp(hipStream_t s, const float* xyz1, int N1, const float* p1, int C1,
            const float* xyz2, int N2, const float* p2, int C2,
            const Layer* Ls, int nL, float* out, int* nnidx, float* nnw,
            float* bufA, float* bufB) {
  int tot = BB * N1;
  k_three_nn<<<(tot + 255) / 256, 256, 0, s>>>(xyz1, xyz2, nnidx, nnw, N1, N2);
  size_t ce = (size_t)tot * (C1 + C2);
  k_concat_interp<<<(unsigned)((ce + 255) / 256), 256, 0, s>>>(
      p1, p2, nnidx, nnw, bufA, N1, N2, C1, C2);
  run_mlp(s, bufA, tot, Ls, nL, bufA, bufB, out);
}

}  // namespace

extern "C" void kernel_launch(void* const* d_in, const int* in_sizes, int n_in,
                              void* d_out, int out_size, void* d_ws,
                              size_t ws_size, hipStream_t stream) {
  (void)in_sizes; (void)n_in; (void)out_size; (void)ws_size;

  const float* xyz_raw = (const float*)d_in[0];    // (B,3,4096)
  const float* pts_raw = (const float*)d_in[1];    // (B,3,4096)

  // ---- parameter unpack (insertion order: per layer W,b,g,be) ----
  int pi = 2;
  auto take = [&](int cin, int cout) {
    Layer L;
    L.W = (const float*)d_in[pi++];
    L.b = (const float*)d_in[pi++];
    L.g = (const float*)d_in[pi++];
    L.be = (const float*)d_in[pi++];
    L.cin = cin;
    L.cout = cout;
    return L;
  };
  Layer sa1[3] = {take(6, 32), take(32, 32), take(32, 64)};
  Layer sa2[3] = {take(67, 64), take(64, 64), take(64, 128)};
  Layer sa3[3] = {take(131, 128), take(128, 128), take(128, 256)};
  Layer sa4[3] = {take(259, 256), take(256, 256), take(256, 512)};
  Layer fp4[2] = {take(768, 256), take(256, 256)};
  Layer fp3[2] = {take(384, 256), take(256, 256)};
  Layer fp2[2] = {take(320, 256), take(256, 128)};
  Layer fp1[3] = {take(131, 128), take(128, 128), take(128, 128)};
  Layer head1[1] = {take(128, 128)};
  Layer conv2;
  conv2.W = (const float*)d_in[pi++];
  conv2.b = (const float*)d_in[pi++];
  conv2.g = nullptr;
  conv2.be = nullptr;
  conv2.cin = 128;
  conv2.cout = 13;

  // ---- workspace bump allocator ----
  char* ws = (char*)d_ws;
  size_t off = 0;
  auto allocB = [&](size_t bytes) -> char* {
    char* p = ws + off;
    off = (off + bytes + 255) & ~(size_t)255;
    return p;
  };
  auto allocF = [&](size_t n) { return (float*)allocB(n * sizeof(float)); };
  auto allocI = [&](size_t n) { return (int*)allocB(n * sizeof(int)); };

  float* x0t = allocF((size_t)BB * NPTS * 3);
  float* p0t = allocF((size_t)BB * NPTS * 3);
  float* x1 = allocF((size_t)BB * 1024 * 3);
  float* x2 = allocF((size_t)BB * 256 * 3);
  float* x3 = allocF((size_t)BB * 64 * 3);
  float* x4 = allocF((size_t)BB * 16 * 3);
  float* p1f = allocF((size_t)BB * 1024 * 64);
  float* p2f = allocF((size_t)BB * 256 * 128);
  float* p3f = allocF((size_t)BB * 64 * 256);
  float* p4f = allocF((size_t)BB * 16 * 512);
  float* p3n = allocF((size_t)BB * 64 * 256);
  float* p2n = allocF((size_t)BB * 256 * 256);
  float* p1n = allocF((size_t)BB * 1024 * 128);
  float* p0n = allocF((size_t)BB * NPTS * 128);
  int* fidx = allocI((size_t)BB * 1024);
  int* gidx = allocI((size_t)BB * 1024 * 32);
  int* nnidx = allocI((size_t)BB * NPTS * 3);
  float* nnw = allocF((size_t)BB * NPTS * 3);
  const size_t BUF = (size_t)BB * 1024 * 32 * 64;  // 16.8M floats max stage
  float* bufA = allocF(BUF);
  float* bufB = allocF(BUF);

  // ---- pipeline ----
  int tot0 = BB * NPTS;
  k_transpose_b3n<<<(tot0 + 255) / 256, 256, 0, stream>>>(xyz_raw, x0t, NPTS);
  k_transpose_b3n<<<(tot0 + 255) / 256, 256, 0, stream>>>(pts_raw, p0t, NPTS);

  run_sa(stream, x0t, NPTS, p0t, 3, 1024, 0.1f, 32, sa1, 3, x1, p1f, fidx,
         gidx, bufA, bufB);
  run_sa(stream, x1, 1024, p1f, 64, 256, 0.2f, 32, sa2, 3, x2, p2f, fidx,
         gidx, bufA, bufB);
  run_sa(stream, x2, 256, p2f, 128, 64, 0.4f, 32, sa3, 3, x3, p3f, fidx,
         gidx, bufA, bufB);
  run_sa(stream, x3, 64, p3f, 256, 16, 0.8f, 32, sa4, 3, x4, p4f, fidx,
         gidx, bufA, bufB);

  run_fp(stream, x3, 64, p3f, 256, x4, 16, p4f, 512, fp4, 2, p3n, nnidx, nnw,
         bufA, bufB);
  run_fp(stream, x2, 256, p2f, 128, x3, 64, p3n, 256, fp3, 2, p2n, nnidx,
         nnw, bufA, bufB);
  run_fp(stream, x1, 1024, p1f, 64, x2, 256, p2n, 256, fp2, 2, p1n, nnidx,
         nnw, bufA, bufB);
  run_fp(stream, x0t, NPTS, p0t, 3, x1, 1024, p1n, 128, fp1, 3, p0n, nnidx,
         nnw, bufA, bufB);

  // head1 (BN+ReLU) then final 128->13 linear into d_out (B,N,13).
  launch_gemm(stream, p0n, head1[0], bufA, tot0, /*bnrelu=*/1);
  launch_gemm(stream, bufA, conv2, (float*)d_out, tot0, /*bnrelu=*/0);
}